// SwarmReasoningNetwork_23476291240576
// MI455X (gfx1250) — compile-verified
//
#include <hip/hip_runtime.h>
#include <hip/hip_bf16.h>

typedef __attribute__((ext_vector_type(2))) float v2f;
typedef __attribute__((ext_vector_type(8))) float v8f;
typedef __attribute__((ext_vector_type(4))) int   v4i;

typedef __attribute__((address_space(1))) v4i gv4i;   // global
typedef __attribute__((address_space(3))) v4i lv4i;   // LDS

#define NN 1024
#define DD 128
#define KP 144   // padded K for the input projection (137 -> 144)

#if __has_builtin(__builtin_amdgcn_global_load_async_to_lds_b128)
#define ASYNC_LDS_OK 1
#endif

__device__ __forceinline__ float gelu_exact(float x) {
    return 0.5f * x * (1.0f + erff(x * 0.70710678118654752f));
}

// 16-byte global->LDS copy: async DMA (ASYNCcnt-tracked, no VGPR round trip)
// when the gfx1250 builtin is available, plain vector copy otherwise.
__device__ __forceinline__ void copy16_to_lds(const float* gsrc, float* ldst) {
#ifdef ASYNC_LDS_OK
    // flat->global: identity; flat LDS address low 32 bits == LDS offset
    __builtin_amdgcn_global_load_async_to_lds_b128(
        (gv4i*)(unsigned long long)gsrc,
        (lv4i*)(unsigned long long)ldst,
        0, 0);
#else
    *(float4*)ldst = *(const float4*)gsrc;
#endif
}

__device__ __forceinline__ void async_wait0() {
#ifdef ASYNC_LDS_OK
#if __has_builtin(__builtin_amdgcn_s_wait_asynccnt)
    __builtin_amdgcn_s_wait_asynccnt(0);
#else
    asm volatile("s_wait_asynccnt 0" ::: "memory");
#endif
#endif
}

// ---------------------------------------------------------------------------
// Zero-pad node_feats (1024x137 -> 1024x144) and Wp (137x128 -> 144x128).
// ---------------------------------------------------------------------------
__global__ __launch_bounds__(256)
void pad_proj_kernel(const float* __restrict__ nf, const float* __restrict__ Wp,
                     float* __restrict__ nfp, float* __restrict__ Wpp)
{
    const int t = blockIdx.x * 256 + threadIdx.x;
    if (t < NN * KP) {
        const int r = t / KP, c = t - r * KP;
        nfp[t] = (c < 137) ? nf[r * 137 + c] : 0.0f;
    }
    if (t < KP * DD) {
        const int r = t / DD;
        Wpp[t] = (r < 137) ? Wp[t] : 0.0f;
    }
}

// ---------------------------------------------------------------------------
// WMMA f32 GEMM: C[M x 128] = A[M x K](lda) @ B[K x LDB] + bias.
// K multiple of 4 (ragged K pre-padded). LDB compile-time so in-loop B
// addresses fold to immediate offsets. One wave = one 16x16 tile.
// Grid: (128/16, M/64), Block: 128 (4 waves -> 4 consecutive M-tiles).
// ---------------------------------------------------------------------------
template <int LDB>
__global__ __launch_bounds__(128)
void wmma_gemm_f32(const float* __restrict__ A, int lda,
                   const float* __restrict__ B,
                   const float* __restrict__ bias,
                   float* __restrict__ C, int ldc,
                   int K)
{
    const int lane = threadIdx.x & 31;
    const int wave = threadIdx.x >> 5;
    const int half = lane >> 4;     // 0: holds K+0/K+1, 1: holds K+2/K+3
    const int l16  = lane & 15;

    const int tm = blockIdx.y * 4 + wave;   // 16-row tile index
    const int tn = blockIdx.x;              // 16-col tile index
    const int row = tm * 16 + l16;
    const int col = tn * 16 + l16;

    const float* __restrict__ Arow = A + (size_t)row * lda;
    const float* __restrict__ Bcol = B + col;

    v8f acc = {};
#pragma unroll 4
    for (int k = 0; k < K; k += 4) {
        const int ka = k + 2 * half;
        const v2f af = *(const v2f*)(Arow + ka);      // 8B contiguous, aligned
        v2f bf;
        bf.x = Bcol[(ka + 0) * LDB];
        bf.y = Bcol[(ka + 1) * LDB];
        // (neg_a, A, neg_b, B, c_mod, C, reuse_a, reuse_b)
        acc = __builtin_amdgcn_wmma_f32_16x16x4_f32(
            false, af, false, bf, (short)0, acc, false, false);
    }

    const float bv = bias ? bias[col] : 0.0f;
#pragma unroll
    for (int r = 0; r < 8; ++r) {
        const int ro = tm * 16 + r + 8 * half;   // C layout: M = r + 8*(lane>=16)
        C[(size_t)ro * ldc + col] = acc[r] + bv;
    }
}

// ---------------------------------------------------------------------------
// Aggregation: agg[i,d] = sum_j adj[i,j] * gelu(aP[i,d] + cP[j,d])
// (bm already folded into aP). Writes right half of hcat (ld 256).
// Block: (128, 4) = 512 threads, 4 rows per block.
// Double-buffered LDS staging of c (async DMA when available) overlaps the
// next chunk's copy with this chunk's 32K (4x128x64) GELU+FMA evaluations.
// ---------------------------------------------------------------------------
#define AGG_TI 4
#define AGG_JC 64

__global__ __launch_bounds__(512)
void agg_kernel(const float* __restrict__ aP,
                const float* __restrict__ cP,
                const float* __restrict__ adj,
                float* __restrict__ hcat)   // 1024 x 256, write cols 128..255
{
    __shared__ float c_sh[2][AGG_JC][DD];
    __shared__ float adj_sh[2][AGG_TI][AGG_JC];

    const int d  = threadIdx.x;              // 0..127
    const int ti = threadIdx.y;              // 0..3
    const int i  = blockIdx.x * AGG_TI + ti;
    const int tid = ti * DD + d;             // 0..511

    const float a_id = aP[i * DD + d];
    float acc = 0.0f;

    // stage chunk (j0) into buffer buf
    auto stage = [&](int buf, int j0) {
        const int base = j0 * DD;            // contiguous 8192 floats
#pragma unroll
        for (int t = tid; t < AGG_JC * DD / 4; t += 512) {
            copy16_to_lds(cP + base + 4 * t, &c_sh[buf][0][0] + 4 * t);
        }
        if (tid < AGG_TI * AGG_JC) {
            adj_sh[buf][tid >> 6][tid & 63] =
                adj[(blockIdx.x * AGG_TI + (tid >> 6)) * NN + j0 + (tid & 63)];
        }
    };

    stage(0, 0);
    async_wait0();
    __syncthreads();

    int cur = 0;
    for (int j0 = 0; j0 < NN; j0 += AGG_JC) {
        const int nxt = cur ^ 1;
        if (j0 + AGG_JC < NN) stage(nxt, j0 + AGG_JC);   // overlap with compute

#pragma unroll 4
        for (int jj = 0; jj < AGG_JC; ++jj) {
            const float x = a_id + c_sh[cur][jj][d];
            acc = fmaf(adj_sh[cur][ti][jj], gelu_exact(x), acc);
        }

        async_wait0();
        __syncthreads();
        cur = nxt;
    }
    hcat[i * 256 + DD + d] = acc;
}

// ---------------------------------------------------------------------------
// x = gelu(new) + h ; LayerNorm(x)*gamma+beta -> hcat left half (and h_out).
// Block: 128 threads (one row), grid: 1024.
// ---------------------------------------------------------------------------
__global__ __launch_bounds__(128)
void update_ln_kernel(const float* __restrict__ newb,   // 1024 x 128
                      float* __restrict__ hcat,          // 1024 x 256
                      const float* __restrict__ gamma,
                      const float* __restrict__ beta,
                      float* __restrict__ h_out)         // may be null
{
    const int i = blockIdx.x;
    const int d = threadIdx.x;

    const float hv = hcat[i * 256 + d];
    const float nv = newb[i * DD + d];
    const float x  = gelu_exact(nv) + hv;

    float s1 = x, s2 = x * x;
#pragma unroll
    for (int m = 16; m; m >>= 1) {
        s1 += __shfl_xor(s1, m, 32);
        s2 += __shfl_xor(s2, m, 32);
    }
    __shared__ float r1[4], r2[4];
    if ((d & 31) == 0) { r1[d >> 5] = s1; r2[d >> 5] = s2; }
    __syncthreads();
    const float S1 = r1[0] + r1[1] + r1[2] + r1[3];
    const float S2 = r2[0] + r2[1] + r2[2] + r2[3];
    const float mu  = S1 * (1.0f / DD);
    const float var = S2 * (1.0f / DD) - mu * mu;
    const float y = (x - mu) * rsqrtf(var + 1e-5f) * gamma[d] + beta[d];

    hcat[i * 256 + d] = y;
    if (h_out) h_out[i * DD + d] = y;
}

// ---------------------------------------------------------------------------
// Output heads: threat = 5*sigmoid(h@Wt+bt); intent = h@Wi+bi; action = h@Wa+ba
// Block: 128 threads (one node), grid: 1024.
// ---------------------------------------------------------------------------
__global__ __launch_bounds__(128)
void heads_kernel(const float* __restrict__ hcat,       // h in left half, ld 256
                  const float* __restrict__ Wt, const float* __restrict__ bt,
                  const float* __restrict__ Wi, const float* __restrict__ bi,
                  const float* __restrict__ Wa, const float* __restrict__ ba,
                  float* __restrict__ threat,
                  float* __restrict__ intent,
                  float* __restrict__ action)
{
    const int i = blockIdx.x;
    const int d = threadIdx.x;
    const float hv = hcat[i * 256 + d];

    float p[11];
    p[0] = hv * Wt[d];
#pragma unroll
    for (int o = 0; o < 5; ++o) p[1 + o] = hv * Wi[d * 5 + o];
#pragma unroll
    for (int o = 0; o < 5; ++o) p[6 + o] = hv * Wa[d * 5 + o];

#pragma unroll
    for (int m = 16; m; m >>= 1) {
#pragma unroll
        for (int q = 0; q < 11; ++q) p[q] += __shfl_xor(p[q], m, 32);
    }
    __shared__ float red[4][11];
    if ((d & 31) == 0) {
#pragma unroll
        for (int q = 0; q < 11; ++q) red[d >> 5][q] = p[q];
    }
    __syncthreads();
    if (d < 11) {
        const float s = red[0][d] + red[1][d] + red[2][d] + red[3][d];
        if (d == 0) {
            threat[i] = 5.0f / (1.0f + expf(-(s + bt[0])));
        } else if (d < 6) {
            intent[i * 5 + (d - 1)] = s + bi[d - 1];
        } else {
            action[i * 5 + (d - 6)] = s + ba[d - 6];
        }
    }
}

// ---------------------------------------------------------------------------
extern "C" void kernel_launch(void* const* d_in, const int* in_sizes, int n_in,
                              void* d_out, int out_size, void* d_ws, size_t ws_size,
                              hipStream_t stream) {
    const float* node_feats = (const float*)d_in[0];   // 1024 x 137
    const float* adj        = (const float*)d_in[1];   // 1024 x 1024
    const float* Wp         = (const float*)d_in[2];   // 137 x 128
    const float* bp         = (const float*)d_in[3];   // 128
    const float* Wm         = (const float*)d_in[4];   // 3 x 256 x 128
    const float* bm         = (const float*)d_in[5];   // 3 x 128
    const float* Wu         = (const float*)d_in[6];   // 3 x 256 x 128
    const float* bu         = (const float*)d_in[7];   // 3 x 128
    const float* gamma      = (const float*)d_in[8];   // 3 x 128
    const float* beta       = (const float*)d_in[9];   // 3 x 128
    const float* Wt         = (const float*)d_in[10];  // 128 x 1
    const float* bt         = (const float*)d_in[11];  // 1
    const float* Wi         = (const float*)d_in[12];  // 128 x 5
    const float* bi         = (const float*)d_in[13];  // 5
    const float* Wa         = (const float*)d_in[14];  // 128 x 5
    const float* ba         = (const float*)d_in[15];  // 5

    float* out      = (float*)d_out;
    float* threat   = out;                       // 1024
    float* intent   = out + 1024;                // 5120
    float* action   = out + 1024 + 5120;         // 5120
    float* h_final  = out + 1024 + 5120 + 5120;  // 1024*128

    float* ws   = (float*)d_ws;
    float* hcat = ws;                 // 1024 x 256 : [h | agg]
    float* aP   = hcat + NN * 256;    // 1024 x 128 : a + bm
    float* cP   = aP + NN * DD;       // 1024 x 128
    float* newb = cP + NN * DD;       // 1024 x 128
    float* nfp  = newb + NN * DD;     // 1024 x 144 (padded node_feats)
    float* Wpp  = nfp + NN * KP;      // 144 x 128  (padded Wp)

    const dim3 gemm_grid(DD / 16, NN / 64);
    const dim3 gemm_block(128);

    // zero-pad projection operands so the GEMM runs guard-free
    pad_proj_kernel<<<dim3((NN * KP + 255) / 256), dim3(256), 0, stream>>>(
        node_feats, Wp, nfp, Wpp);

    // h = node_feats @ Wp + bp  -> hcat[:, :128]
    wmma_gemm_f32<DD><<<gemm_grid, gemm_block, 0, stream>>>(
        nfp, KP, Wpp, bp, hcat, 256, KP);

    for (int l = 0; l < 3; ++l) {
        const float* Wm_l = Wm + (size_t)l * 2 * DD * DD;
        const float* Wu_l = Wu + (size_t)l * 2 * DD * DD;

        // aP = h @ Wm[:D] + bm   (bm folded here)
        wmma_gemm_f32<DD><<<gemm_grid, gemm_block, 0, stream>>>(
            hcat, 256, Wm_l, bm + l * DD, aP, DD, DD);
        // cP = h @ Wm[D:]
        wmma_gemm_f32<DD><<<gemm_grid, gemm_block, 0, stream>>>(
            hcat, 256, Wm_l + DD * DD, nullptr, cP, DD, DD);

        // agg -> hcat[:, 128:]
        agg_kernel<<<dim3(NN / AGG_TI), dim3(DD, AGG_TI), 0, stream>>>(
            aP, cP, adj, hcat);

        // new = [h|agg] @ Wu + bu   (K = 256)
        wmma_gemm_f32<DD><<<gemm_grid, gemm_block, 0, stream>>>(
            hcat, 256, Wu_l, bu + l * DD, newb, DD, 2 * DD);

        // h = LN(gelu(new) + h) * gamma + beta
        update_ln_kernel<<<dim3(NN), dim3(DD), 0, stream>>>(
            newb, hcat, gamma + l * DD, beta + l * DD,
            (l == 2) ? h_final : nullptr);
    }

    heads_kernel<<<dim3(NN), dim3(DD), 0, stream>>>(
        hcat, Wt, bt, Wi, bi, Wa, ba, threat, intent, action);
}